// GATClassifier_38783554683006
// MI455X (gfx1250) — compile-verified
//
#include <hip/hip_runtime.h>
#include <hip/hip_bf16.h>

// ---------------- constants matching the reference ----------------
#define Dd     256      // H*C
#define Hh     4
#define Cc     64
#define OUTC   10
#define NEG_SLOPE 0.2f
#define BN_EPS    1e-5f

typedef float v2f __attribute__((ext_vector_type(2)));
typedef float v8f __attribute__((ext_vector_type(8)));

// order-preserving float<->uint encoding for atomic max on floats
__device__ __forceinline__ unsigned f2ord(float f) {
  unsigned b = __float_as_uint(f);
  return (b & 0x80000000u) ? ~b : (b | 0x80000000u);
}
__device__ __forceinline__ float ord2f(unsigned u) {
  unsigned b = (u & 0x80000000u) ? (u & 0x7FFFFFFFu) : ~u;
  return __uint_as_float(b);
}
__device__ __forceinline__ float lrelu(float x) { return x > 0.f ? x : NEG_SLOPE * x; }

// ---------------- fp32 WMMA GEMM:  C[M,256] = A[M,K] @ W[K,256] ----------------
// block = 128 threads (4 waves). Block computes 16 rows x 64 cols.
// A 16xK tile staged in LDS via GLOBAL_LOAD_ASYNC_TO_LDS_B128 (ASYNCcnt path,
// no VGPR staging), row stride K+4 to avoid LDS bank conflicts.
// Each wave: one 16x16 tile, K-loop of V_WMMA_F32_16X16X4_F32.
__global__ __launch_bounds__(128)
void gemm_wmma_kernel(const float* __restrict__ A, const float* __restrict__ W,
                      float* __restrict__ C, int K) {
  extern __shared__ float ldsA[];           // 16 * (K+4)
  const int ldk  = K + 4;
  const int row0 = blockIdx.x * 16;
  const int col0 = blockIdx.y * 64;
  const int tid  = threadIdx.x;

  // async 16B-per-lane copy of the 16xK A tile into LDS.
  // LDS byte offset = addr[31:0] of the generic LDS pointer (HW adds LDS_BASE).
  const int totalA = 16 * K;
  for (int i = tid * 4; i < totalA; i += 128 * 4) {
    int r = i / K, c = i % K;                         // K % 4 == 0 -> stays in row
    unsigned     ldsOff = (unsigned)(uintptr_t)&ldsA[r * ldk + c];
    const float* gsrc   = &A[(size_t)(row0 + r) * K + c];
    asm volatile("global_load_async_to_lds_b128 %0, %1, off"
                 :: "v"(ldsOff), "v"(gsrc)
                 : "memory");
  }
  asm volatile("s_wait_asynccnt 0" ::: "memory");
  __syncthreads();

  const int wave = tid >> 5;
  const int lane = tid & 31;
  const int m    = lane & 15;               // A row within tile
  const int kk   = (lane >> 4) << 1;        // K sub-offset: lanes 0-15 -> K{0,1}; 16-31 -> K{2,3}
  const int col  = col0 + wave * 16 + (lane & 15);

  v8f acc = {};
  for (int k0 = 0; k0 < K; k0 += 4) {
    v2f a, b;
    a.x = ldsA[m * ldk + k0 + kk];
    a.y = ldsA[m * ldk + k0 + kk + 1];
    b.x = W[(size_t)(k0 + kk)     * Dd + col];
    b.y = W[(size_t)(k0 + kk + 1) * Dd + col];
    acc = __builtin_amdgcn_wmma_f32_16x16x4_f32(false, a, false, b, (short)0, acc,
                                                false, false);
  }
  // C/D layout: VGPR r -> rows r (lanes 0-15) / r+8 (lanes 16-31)
  const int rbase = row0 + ((lane >> 4) << 3);
#pragma unroll
  for (int r = 0; r < 8; ++r)
    C[(size_t)(rbase + r) * Dd + col] = acc[r];
}

// ---------------- per-node attention dot products ----------------
// alpha_s[n,h] = sum_c h[n,h,c]*a_src[h,c];  one block (256 thr) per node
__global__ __launch_bounds__(256)
void alpha_kernel(const float* __restrict__ h, const float* __restrict__ asrc,
                  const float* __restrict__ adst, float* __restrict__ alpha_s,
                  float* __restrict__ alpha_d) {
  __shared__ float ss[256], sd[256];
  const int n = blockIdx.x, t = threadIdx.x;
  float v = h[(size_t)n * Dd + t];
  ss[t] = v * asrc[t];
  sd[t] = v * adst[t];
  __syncthreads();
  for (int off = 32; off > 0; off >>= 1) {
    if ((t & 63) < off) { ss[t] += ss[t + off]; sd[t] += sd[t + off]; }
    __syncthreads();
  }
  if ((t & 63) == 0) {
    int head = t >> 6;
    alpha_s[n * Hh + head] = ss[t];
    alpha_d[n * Hh + head] = sd[t];
  }
}

__device__ __forceinline__ void edge_sd(const int* __restrict__ ei, int e, int E_,
                                        int& s, int& d) {
  if (e < E_) { s = ei[e]; d = ei[E_ + e]; }
  else        { s = d = e - E_; }           // self loops appended
}

// pass 1: segment max over dst (ordered-uint atomic max)
__global__ void edge_max_kernel(const float* __restrict__ as, const float* __restrict__ ad,
                                const int* __restrict__ ei, int E_, int ET,
                                unsigned* __restrict__ maxv) {
  int idx = blockIdx.x * blockDim.x + threadIdx.x;
  if (idx >= ET * Hh) return;
  int e = idx >> 2, hd = idx & 3;
  int s, d; edge_sd(ei, e, E_, s, d);
  float ev = lrelu(as[s * Hh + hd] + ad[d * Hh + hd]);
  atomicMax(&maxv[d * Hh + hd], f2ord(ev));
}

// pass 2: segment sum of exp(e - max)
__global__ void edge_sum_kernel(const float* __restrict__ as, const float* __restrict__ ad,
                                const int* __restrict__ ei, int E_, int ET,
                                const unsigned* __restrict__ maxv, float* __restrict__ sumv) {
  int idx = blockIdx.x * blockDim.x + threadIdx.x;
  if (idx >= ET * Hh) return;
  int e = idx >> 2, hd = idx & 3;
  int s, d; edge_sd(ei, e, E_, s, d);
  float ev = lrelu(as[s * Hh + hd] + ad[d * Hh + hd]);
  atomicAdd(&sumv[d * Hh + hd], __expf(ev - ord2f(maxv[d * Hh + hd])));
}

// pass 3: out[dst] += alpha * h[src]  (one block of 256 per edge, fully coalesced)
__global__ __launch_bounds__(256)
void edge_agg_kernel(const float* __restrict__ h, const float* __restrict__ as,
                     const float* __restrict__ ad, const unsigned* __restrict__ maxv,
                     const float* __restrict__ sumv, const int* __restrict__ ei, int E_,
                     float* __restrict__ out) {
  const int e = blockIdx.x, t = threadIdx.x, hd = t >> 6;
  int s, d; edge_sd(ei, e, E_, s, d);
  float ev    = lrelu(as[s * Hh + hd] + ad[d * Hh + hd]);
  float m     = ord2f(maxv[d * Hh + hd]);
  float ssum  = sumv[d * Hh + hd];
  float alpha = __expf(ev - m) / (ssum + 1e-16f);
  atomicAdd(&out[(size_t)d * Dd + t], h[(size_t)s * Dd + t] * alpha);
}

__global__ void bias_relu_kernel(float* __restrict__ out, const float* __restrict__ bias,
                                 int total) {
  int idx = blockIdx.x * blockDim.x + threadIdx.x;
  if (idx < total) out[idx] = fmaxf(out[idx] + bias[idx & (Dd - 1)], 0.f);
}

// ---------------- pooling / BN / head ----------------
__global__ void pool_cnt_kernel(const int* __restrict__ batch, float* __restrict__ cnt, int Nn) {
  int n = blockIdx.x * blockDim.x + threadIdx.x;
  if (n < Nn) atomicAdd(&cnt[batch[n]], 1.f);
}
__global__ void pool_accum_kernel(const float* __restrict__ node, const int* __restrict__ batch,
                                  float* __restrict__ pooled, int Nn) {
  int idx = blockIdx.x * blockDim.x + threadIdx.x;
  if (idx < Nn * Dd) {
    int n = idx >> 8;
    atomicAdd(&pooled[batch[n] * Dd + (idx & (Dd - 1))], node[idx]);
  }
}
__global__ void pool_div_kernel(float* __restrict__ pooled, const float* __restrict__ cnt, int G_) {
  int idx = blockIdx.x * blockDim.x + threadIdx.x;
  if (idx < G_ * Dd) pooled[idx] /= fmaxf(cnt[idx >> 8], 1.f);
}
__global__ __launch_bounds__(256)
void bn_stats_kernel(const float* __restrict__ pooled, float* __restrict__ mu,
                     float* __restrict__ var, int G_) {
  int d = threadIdx.x;
  float s = 0.f, sq = 0.f;
  for (int g = 0; g < G_; ++g) { float v = pooled[g * Dd + d]; s += v; sq += v * v; }
  float m = s / (float)G_;
  mu[d]  = m;
  var[d] = sq / (float)G_ - m * m;
}
// graph_emb + logits + softmax; one block (256 thr) per graph
__global__ __launch_bounds__(256)
void graph_final_kernel(const float* __restrict__ pooled, const float* __restrict__ mu,
                        const float* __restrict__ var, const float* __restrict__ gamma,
                        const float* __restrict__ beta, const float* __restrict__ mlp_w,
                        const float* __restrict__ mlp_b, float* __restrict__ out,
                        int G_, int Nn) {
  __shared__ float sge[256];
  __shared__ float slog[16];
  const int g = blockIdx.x, t = threadIdx.x;
  float ge = (pooled[g * Dd + t] - mu[t]) * rsqrtf(var[t] + BN_EPS) * gamma[t] + beta[t];
  out[(size_t)2 * G_ * OUTC + (size_t)Nn * Dd + (size_t)g * Dd + t] = ge;  // graph_emb
  sge[t] = ge;
  __syncthreads();
  if (t < OUTC) {
    float s = mlp_b[t];
    for (int d2 = 0; d2 < Dd; ++d2) s += sge[d2] * mlp_w[d2 * OUTC + t];
    slog[t] = s;
    out[g * OUTC + t] = s;                                                // logits
  }
  __syncthreads();
  if (t == 0) {
    float mx = slog[0];
    for (int i = 1; i < OUTC; ++i) mx = fmaxf(mx, slog[i]);
    float ex[OUTC], s = 0.f;
    for (int i = 0; i < OUTC; ++i) { ex[i] = __expf(slog[i] - mx); s += ex[i]; }
    for (int i = 0; i < OUTC; ++i) out[G_ * OUTC + g * OUTC + i] = ex[i] / s;  // probs
  }
}

// ---------------- host side ----------------
extern "C" void kernel_launch(void* const* d_in, const int* in_sizes, int n_in,
                              void* d_out, int out_size, void* d_ws, size_t ws_size,
                              hipStream_t stream) {
  const float* x     = (const float*)d_in[0];
  const int*   ei    = (const int*)d_in[1];
  const int*   batch = (const int*)d_in[2];
  const float* bn_gamma = (const float*)d_in[15];
  const float* bn_beta  = (const float*)d_in[16];
  const float* mlp_w    = (const float*)d_in[17];
  const float* mlp_b    = (const float*)d_in[18];

  const int N    = in_sizes[2];          // batch is [N]
  const int E    = in_sizes[1] / 2;      // edge_index is [2,E]
  const int F_IN = in_sizes[0] / N;
  const int ET   = E + N;                // with self loops
  const int G    = (out_size - N * Dd) / (2 * OUTC + Dd);

  float* ws       = (float*)d_ws;
  float* bufA     = ws;                          // h   : N*Dd
  float* bufB     = bufA + (size_t)N * Dd;       // ping: N*Dd
  float* alpha_s  = bufB + (size_t)N * Dd;       // N*Hh
  float* alpha_d  = alpha_s + (size_t)N * Hh;    // N*Hh
  unsigned* maxv  = (unsigned*)(alpha_d + (size_t)N * Hh);   // N*Hh
  float* sumv     = (float*)(maxv + (size_t)N * Hh);         // N*Hh
  float* pooled   = sumv + (size_t)N * Hh;       // G*Dd
  float* cnt      = pooled + (size_t)G * Dd;     // G
  float* mu       = cnt + G;                     // Dd
  float* var      = mu + Dd;                     // Dd

  float* node_emb = (float*)d_out + 2 * G * OUTC;  // layer-2 output lives in d_out

  const int totND    = N * Dd;
  const int gridEH   = (ET * Hh + 255) / 256;
  const int gridND   = (totND + 255) / 256;

  for (int l = 0; l < 3; ++l) {
    const int    K    = (l == 0) ? F_IN : Dd;
    const float* xin  = (l == 0) ? x : bufB;
    float*       hbuf = bufA;
    float*       out  = (l == 2) ? node_emb : bufB;
    const float* W    = (const float*)d_in[3 + 4 * l];
    const float* asrc = (const float*)d_in[4 + 4 * l];
    const float* adst = (const float*)d_in[5 + 4 * l];
    const float* bias = (const float*)d_in[6 + 4 * l];

    dim3 gg(N / 16, Dd / 64);
    size_t smem = (size_t)16 * (K + 4) * sizeof(float);
    gemm_wmma_kernel<<<gg, 128, smem, stream>>>(xin, W, hbuf, K);

    hipMemsetAsync(out,  0, (size_t)totND * 4, stream);   // after GEMM consumed `out` as input
    hipMemsetAsync(maxv, 0, (size_t)N * Hh * 4, stream);  // 0u == ordered floor
    hipMemsetAsync(sumv, 0, (size_t)N * Hh * 4, stream);

    alpha_kernel<<<N, 256, 0, stream>>>(hbuf, asrc, adst, alpha_s, alpha_d);
    edge_max_kernel<<<gridEH, 256, 0, stream>>>(alpha_s, alpha_d, ei, E, ET, maxv);
    edge_sum_kernel<<<gridEH, 256, 0, stream>>>(alpha_s, alpha_d, ei, E, ET, maxv, sumv);
    edge_agg_kernel<<<ET, 256, 0, stream>>>(hbuf, alpha_s, alpha_d, maxv, sumv, ei, E, out);
    bias_relu_kernel<<<gridND, 256, 0, stream>>>(out, bias, totND);
  }

  // readout
  hipMemsetAsync(pooled, 0, (size_t)G * Dd * 4, stream);
  hipMemsetAsync(cnt,    0, (size_t)G * 4, stream);
  pool_cnt_kernel  <<<(N + 255) / 256, 256, 0, stream>>>(batch, cnt, N);
  pool_accum_kernel<<<gridND, 256, 0, stream>>>(node_emb, batch, pooled, N);
  pool_div_kernel  <<<(G * Dd + 255) / 256, 256, 0, stream>>>(pooled, cnt, G);
  bn_stats_kernel  <<<1, 256, 0, stream>>>(pooled, mu, var, G);
  graph_final_kernel<<<G, 256, 0, stream>>>(pooled, mu, var, bn_gamma, bn_beta,
                                            mlp_w, mlp_b, (float*)d_out, G, N);
}